// EBP_binaryNet_53532472377586
// MI455X (gfx1250) — compile-verified
//
#include <hip/hip_runtime.h>
#include <hip/hip_bf16.h>

// ---------------------------------------------------------------------------
// EBP binaryNet forward on MI455X (gfx1250, wave32).
// Bound by the 1.07 GB xcov_4 output write (~46us @ 23.3 TB/s HBM), so GEMMs
// stay full fp32 via V_WMMA_F32_16X16X4_F32 (precision costs nothing here).
// ---------------------------------------------------------------------------

typedef __attribute__((ext_vector_type(2))) float v2f;
typedef __attribute__((ext_vector_type(4))) float v4f;
typedef __attribute__((ext_vector_type(8))) float v8f;

#define SQ2PI 0.79788456f

// ---------------- elementwise m = 2*sigmoid(w) - 1 -------------------------
__global__ void map_sigmoid(const float* __restrict__ w, float* __restrict__ m, int n) {
  int i = blockIdx.x * blockDim.x + threadIdx.x;
  if (i < n) {
    float v = w[i];
    m[i] = 2.0f / (1.0f + __expf(-v)) - 1.0f;
  }
}

// ---------------- s[col] = sum_k (1 - m[k,col]^2) --------------------------
__global__ void __launch_bounds__(256) colsum_1m2(const float* __restrict__ m,
                                                  float* __restrict__ s,
                                                  int K, int N) {
  int col = blockIdx.x;
  float acc = 0.0f;
  for (int k = threadIdx.x; k < K; k += blockDim.x) {
    float v = m[k * N + col];
    acc += 1.0f - v * v;
  }
  __shared__ float red[256];
  red[threadIdx.x] = acc;
  __syncthreads();
  for (int st = 128; st > 0; st >>= 1) {
    if (threadIdx.x < st) red[threadIdx.x] += red[threadIdx.x + st];
    __syncthreads();
  }
  if (threadIdx.x == 0) s[col] = red[0];
}

// ---------------- layer 1: x1 = tanh(SQ2PI*(x@m0+th0)/sqrt(s0)) ------------
// One wave per 16x16 output tile. K = 784 (multiple of 4).
// A frag (16x4 f32, ISA 7.12.2): lane&15 = row, v0 holds K = 2*(lane>>4),
// v1 holds K = 2*(lane>>4)+1.  B frag symmetric (lane&15 = col).
// C layout: element r of v8f at lane l -> row = r + 8*(l>>4), col = l&15.
__global__ void __launch_bounds__(32) layer1_wmma(
    const float* __restrict__ X, const float* __restrict__ M0,
    const float* __restrict__ S0, const float* __restrict__ TH0,
    float* __restrict__ X1, float* __restrict__ C1) {
  const int K = 784, N = 1024;
  int tm = blockIdx.x >> 6;   // 16 row tiles (M = 256)
  int tn = blockIdx.x & 63;   // 64 col tiles (N = 1024)
  int lane = threadIdx.x;
  int hf = lane >> 4;
  int lm = lane & 15;
  int arow = tm * 16 + lm;
  int bcol = tn * 16 + lm;

  v8f acc = {};
  for (int k0 = 0; k0 < K; k0 += 4) {
    int ka = k0 + 2 * hf;
    v2f a, b;
    a.x = X[arow * K + ka];
    a.y = X[arow * K + ka + 1];
    b.x = M0[ka * N + bcol];
    b.y = M0[(ka + 1) * N + bcol];
    acc = __builtin_amdgcn_wmma_f32_16x16x4_f32(false, a, false, b,
                                                (short)0, acc, false, false);
  }

  int col = tn * 16 + lm;
  float th = TH0[col];
  float rs = rsqrtf(S0[col]);
#pragma unroll
  for (int r = 0; r < 8; ++r) {
    int row = tm * 16 + r + 8 * hf;
    float h  = SQ2PI * (acc[r] + th) * rs;
    float xv = tanhf(h);
    X1[row * N + col] = xv;
    C1[row * N + col] = 1.0f - xv * xv;
  }
}

// ---------------- fused EBP layer: two GEMMs + activation ------------------
// acc_h = xbar @ m ; acc_s = cdiag @ m^2  (b^2 squared in-register, so the
// weight tile is loaded once and feeds both WMMA accumulators).
__global__ void __launch_bounds__(32) ebp_layer_wmma(
    const float* __restrict__ XB, const float* __restrict__ CD,
    const float* __restrict__ Mw, const float* __restrict__ S,
    const float* __restrict__ TH,
    float* __restrict__ XN, float* __restrict__ CN) {
  const int K = 1024, N = 1024;
  int tm = blockIdx.x >> 6;
  int tn = blockIdx.x & 63;
  int lane = threadIdx.x;
  int hf = lane >> 4;
  int lm = lane & 15;
  int arow = tm * 16 + lm;
  int bcol = tn * 16 + lm;

  v8f acc_h = {}, acc_s = {};
  for (int k0 = 0; k0 < K; k0 += 4) {
    int ka = k0 + 2 * hf;
    v2f ax, ac, b, b2;
    ax.x = XB[arow * K + ka];
    ax.y = XB[arow * K + ka + 1];
    ac.x = CD[arow * K + ka];
    ac.y = CD[arow * K + ka + 1];
    b.x = Mw[ka * N + bcol];
    b.y = Mw[(ka + 1) * N + bcol];
    b2.x = b.x * b.x;
    b2.y = b.y * b.y;
    acc_h = __builtin_amdgcn_wmma_f32_16x16x4_f32(false, ax, false, b,
                                                  (short)0, acc_h, false, false);
    acc_s = __builtin_amdgcn_wmma_f32_16x16x4_f32(false, ac, false, b2,
                                                  (short)0, acc_s, false, false);
  }

  int col = tn * 16 + lm;
  float th = TH[col];
  float sc = S[col];
#pragma unroll
  for (int r = 0; r < 8; ++r) {
    int row = tm * 16 + r + 8 * hf;
    float diagsig = acc_s[r] + sc;
    float hbar    = acc_h[r] + th;
    float h  = SQ2PI * hbar * rsqrtf(diagsig);
    float xv = tanhf(h);
    XN[row * N + col] = xv;
    CN[row * N + col] = 1.0f - xv * xv;
  }
}

// ---------------- last layer (D_OUT = 10, tiny) ----------------------------
__global__ void last_layer(const float* __restrict__ X4, const float* __restrict__ C4,
                           const float* __restrict__ ML, const float* __restrict__ SL,
                           const float* __restrict__ THL,
                           float* __restrict__ hlastbar, float* __restrict__ DSL) {
  int idx = blockIdx.x * blockDim.x + threadIdx.x;
  if (idx >= 256 * 10) return;
  int row = idx / 10;
  int col = idx % 10;
  float ah = 0.0f, as = 0.0f;
  for (int k = 0; k < 1024; ++k) {
    float mv = ML[k * 10 + col];
    ah += X4[row * 1024 + k] * mv;
    as += C4[row * 1024 + k] * mv * mv;
  }
  hlastbar[idx] = ah + THL[col];
  DSL[idx]     = as + SL[col];
}

// ---------------- log-softmax, L2Loss, fraction_correct --------------------
__global__ void __launch_bounds__(256) finalize(
    const float* __restrict__ hlastbar, const float* __restrict__ DSL,
    const int* __restrict__ target, float* __restrict__ logprobs,
    float* __restrict__ out_scalars) {
  int row = threadIdx.x;  // 256 rows, one per thread
  float h[10], d[10];
#pragma unroll
  for (int j = 0; j < 10; ++j) {
    h[j] = hlastbar[row * 10 + j];
    d[j] = DSL[row * 10 + j];
  }
  // log_softmax + argmax (first-max, matching jnp.argmax)
  float mx = h[0];
  int am = 0;
#pragma unroll
  for (int j = 1; j < 10; ++j)
    if (h[j] > mx) { mx = h[j]; am = j; }
  float se = 0.0f;
#pragma unroll
  for (int j = 0; j < 10; ++j) se += __expf(h[j] - mx);
  float lse = mx + __logf(se);
#pragma unroll
  for (int j = 0; j < 10; ++j) logprobs[row * 10 + j] = h[j] - lse;

  int tgt = target[row];
  float isit = 2.0f * (float)tgt - 1.0f;
  float is2 = isit * isit;
  float lsum = 0.0f;
#pragma unroll
  for (int j = 0; j < 10; ++j) {
    float hl  = SQ2PI * h[j] * rsqrtf(d[j]);
    float s   = sqrtf(2.0f * d[j]) + 1.0f;
    float th2 = is2 - __expf(-(h[j] * h[j]) / s) / s;
    lsum += is2 - 2.0f * isit * tanhf(hl) + th2;
  }
  float corr = (am == tgt) ? 1.0f : 0.0f;

  __shared__ float rl[256], rc[256];
  rl[row] = lsum;
  rc[row] = corr;
  __syncthreads();
  for (int st = 128; st > 0; st >>= 1) {
    if (row < st) { rl[row] += rl[row + st]; rc[row] += rc[row + st]; }
    __syncthreads();
  }
  if (row == 0) {
    out_scalars[0] = rl[0];
    out_scalars[1] = rc[0] / 256.0f;
  }
}

// ---------------- xcov_4 = eye(H) * c4[:,None,:]  (1.07 GB stream) ---------
// float4 non-temporal stores: this is the HBM-bandwidth floor of the problem.
__global__ void __launch_bounds__(256) write_xcov(const float* __restrict__ C4,
                                                  float* __restrict__ out) {
  // f indexes float4 quads over [256, 1024, 1024]; 2^18 quads per sample.
  unsigned int f = blockIdx.x * 256u + threadIdx.x;  // < 2^26
  int m   = (int)(f >> 18);
  int rem = (int)(f & 262143u);
  int i   = rem >> 8;     // middle index (row of the HxH eye)
  int j4  = rem & 255;    // quad along last dim
  v4f v = {0.0f, 0.0f, 0.0f, 0.0f};
  if ((i >> 2) == j4) v[i & 3] = C4[m * 1024 + i];
  __builtin_nontemporal_store(v, (v4f*)out + f);
}

// ---------------------------------------------------------------------------
extern "C" void kernel_launch(void* const* d_in, const int* in_sizes, int n_in,
                              void* d_out, int out_size, void* d_ws, size_t ws_size,
                              hipStream_t stream) {
  const float* x      = (const float*)d_in[0];
  const int*   target = (const int*)d_in[1];
  const float* w0     = (const float*)d_in[2];
  const float* w1     = (const float*)d_in[3];
  const float* w2     = (const float*)d_in[4];
  const float* w3     = (const float*)d_in[5];
  // d_in[6] = w4 : computed-but-unused in the reference forward
  const float* wlast  = (const float*)d_in[7];
  const float* th0    = (const float*)d_in[8];
  const float* th1    = (const float*)d_in[9];
  const float* th2    = (const float*)d_in[10];
  const float* th3    = (const float*)d_in[11];
  // d_in[12] = th4 : unused
  const float* thlast = (const float*)d_in[13];

  float* out = (float*)d_out;
  float* ws  = (float*)d_ws;

  // workspace layout (floats) — ~24.3 MB total
  size_t o = 0;
  float* m0    = ws + o; o += 802816;   // 784*1024
  float* m1    = ws + o; o += 1048576;
  float* m2    = ws + o; o += 1048576;
  float* m3    = ws + o; o += 1048576;
  float* mlast = ws + o; o += 10240;
  float* s0    = ws + o; o += 1024;
  float* s1    = ws + o; o += 1024;
  float* s2    = ws + o; o += 1024;
  float* s3    = ws + o; o += 1024;
  float* slast = ws + o; o += 16;
  float* x1    = ws + o; o += 262144;   // 256*1024 each
  float* c1    = ws + o; o += 262144;
  float* x2    = ws + o; o += 262144;
  float* c2    = ws + o; o += 262144;
  float* x3    = ws + o; o += 262144;
  float* c3    = ws + o; o += 262144;
  float* x4    = ws + o; o += 262144;
  float* c4    = ws + o; o += 262144;
  float* dsl   = ws + o; o += 2560;

  // 1) weight transforms
  map_sigmoid<<<(802816 + 255) / 256, 256, 0, stream>>>(w0, m0, 802816);
  map_sigmoid<<<(1048576 + 255) / 256, 256, 0, stream>>>(w1, m1, 1048576);
  map_sigmoid<<<(1048576 + 255) / 256, 256, 0, stream>>>(w2, m2, 1048576);
  map_sigmoid<<<(1048576 + 255) / 256, 256, 0, stream>>>(w3, m3, 1048576);
  map_sigmoid<<<(10240 + 255) / 256, 256, 0, stream>>>(wlast, mlast, 10240);

  // 2) column sums of (1 - m^2)
  colsum_1m2<<<1024, 256, 0, stream>>>(m0, s0, 784, 1024);
  colsum_1m2<<<1024, 256, 0, stream>>>(m1, s1, 1024, 1024);
  colsum_1m2<<<1024, 256, 0, stream>>>(m2, s2, 1024, 1024);
  colsum_1m2<<<1024, 256, 0, stream>>>(m3, s3, 1024, 1024);
  colsum_1m2<<<10, 256, 0, stream>>>(mlast, slast, 1024, 10);

  // 3) fused WMMA layers (16x16 tiles, 1024 waves each)
  layer1_wmma<<<1024, 32, 0, stream>>>(x, m0, s0, th0, x1, c1);
  ebp_layer_wmma<<<1024, 32, 0, stream>>>(x1, c1, m1, s1, th1, x2, c2);
  ebp_layer_wmma<<<1024, 32, 0, stream>>>(x2, c2, m2, s2, th2, x3, c3);
  // faithful reference bug: layer 4 consumes c2 (not c3)
  ebp_layer_wmma<<<1024, 32, 0, stream>>>(x3, c2, m3, s3, th3, x4, c4);

  // 4) last layer -> hlastbar (d_out[0:2560]) and diagsiglast (ws)
  last_layer<<<10, 256, 0, stream>>>(x4, c4, mlast, slast, thlast, out, dsl);

  // 5) log-softmax / loss / accuracy
  //    d_out layout: hlastbar[2560] | logprobs[2560] | xcov_4[268435456] | L2Loss | frac
  finalize<<<1, 256, 0, stream>>>(out, dsl, target, out + 2560,
                                  out + (size_t)2560 + 2560 + 268435456ull);

  // 6) stream the 1.07 GB diagonal covariance tensor (bandwidth floor)
  write_xcov<<<262144, 256, 0, stream>>>(c4, out + 5120);
}